// GATLayer_22488448762007
// MI455X (gfx1250) — compile-verified
//
#include <hip/hip_runtime.h>
#include <math.h>

typedef float v2f __attribute__((ext_vector_type(2)));
typedef float v8f __attribute__((ext_vector_type(8)));

#define INDIM 128
#define OUTDIM 64
#define NEG_SLOPE 0.2f

// ---- monotone float <-> uint encoding for atomic max over floats ----
__device__ __forceinline__ unsigned enc_f32(float f) {
  unsigned u = __float_as_uint(f);
  return (u & 0x80000000u) ? ~u : (u | 0x80000000u);
}
__device__ __forceinline__ float dec_f32(unsigned k) {
  unsigned u = (k & 0x80000000u) ? (k ^ 0x80000000u) : ~k;
  return __uint_as_float(u);
}

__device__ __forceinline__ void atomicAddF32(float* addr, float v) {
  __hip_atomic_fetch_add(addr, v, __ATOMIC_RELAXED, __HIP_MEMORY_SCOPE_AGENT);
}

// ============================================================
// Kernel 1: H = X * W^T  via V_WMMA_F32_16X16X4_F32 (wave32).
// Each wave computes a 16-row x 64-col strip of H (4 16x16 tiles).
// W (64x128 = 32KB) staged in LDS once per block.
// ============================================================
__global__ __launch_bounds__(256) void gat_gemm(const float* __restrict__ X,
                                                const float* __restrict__ W,
                                                float* __restrict__ H,
                                                int nRowTiles) {
  __shared__ float sW[OUTDIM * INDIM];
  for (int i = threadIdx.x; i < OUTDIM * INDIM; i += 256) sW[i] = W[i];
  __syncthreads();

  const int wave = threadIdx.x >> 5;
  const int lane = threadIdx.x & 31;
  const int rowTile = blockIdx.x * 8 + wave;
  if (rowTile >= nRowTiles) return;   // uniform per-wave: EXEC stays all-1s for WMMA

  const int m = lane & 15;
  const int khalf = (lane >> 4) << 1;            // 0 or 2 (A/B K-sublane split)
  const float* xrow = X + (size_t)(rowTile * 16 + m) * INDIM;

  v8f c0 = {}, c1 = {}, c2 = {}, c3 = {};
#pragma unroll 4
  for (int k = 0; k < INDIM; k += 4) {
    // A fragment: lane holds X[m][k+khalf], X[m][k+khalf+1]
    v2f a = *(const v2f*)(xrow + k + khalf);
    // B fragment for col-tile t: lane holds W[t*16 + (lane&15)][k+khalf .. +1]
    const float* wb = &sW[(lane & 15) * INDIM + k + khalf];
    v2f b0 = *(const v2f*)(wb + 0 * 16 * INDIM);
    v2f b1 = *(const v2f*)(wb + 1 * 16 * INDIM);
    v2f b2 = *(const v2f*)(wb + 2 * 16 * INDIM);
    v2f b3 = *(const v2f*)(wb + 3 * 16 * INDIM);
    c0 = __builtin_amdgcn_wmma_f32_16x16x4_f32(false, a, false, b0, (short)0, c0, false, false);
    c1 = __builtin_amdgcn_wmma_f32_16x16x4_f32(false, a, false, b1, (short)0, c1, false, false);
    c2 = __builtin_amdgcn_wmma_f32_16x16x4_f32(false, a, false, b2, (short)0, c2, false, false);
    c3 = __builtin_amdgcn_wmma_f32_16x16x4_f32(false, a, false, b3, (short)0, c3, false, false);
  }

  // C/D layout: VGPR r, lanes 0-15: M=r, N=lane; lanes 16-31: M=r+8, N=lane-16
  const int mo = (lane >> 4) << 3;   // 0 or 8
  const int no = lane & 15;
  float* hb = H + (size_t)rowTile * 16 * OUTDIM;
#pragma unroll
  for (int r = 0; r < 8; ++r) {
    float* row = hb + (size_t)(r + mo) * OUTDIM + no;
    row[0]  = c0[r];
    row[16] = c1[r];
    row[32] = c2[r];
    row[48] = c3[r];
  }
}

// ============================================================
// Kernel 2: per-node scores s_src/s_dst = H[n] . a, zero denom & out, init gmax
// ============================================================
__global__ __launch_bounds__(256) void gat_scores(const float* __restrict__ H,
                                                  const float* __restrict__ a_src,
                                                  const float* __restrict__ a_dst,
                                                  float* __restrict__ s_src,
                                                  float* __restrict__ s_dst,
                                                  float* __restrict__ denom,
                                                  float* __restrict__ out,
                                                  unsigned* __restrict__ gmax,
                                                  int N) {
  int n = blockIdx.x * blockDim.x + threadIdx.x;
  if (n == 0) *gmax = 0u;   // minimal key in monotone encoding
  if (n >= N) return;
  const float* h = H + (size_t)n * OUTDIM;
  float* o = out + (size_t)n * OUTDIM;
  float ss = 0.f, sd = 0.f;
#pragma unroll
  for (int d = 0; d < OUTDIM; ++d) {
    float v = h[d];
    ss = fmaf(v, a_src[d], ss);
    sd = fmaf(v, a_dst[d], sd);
    o[d] = 0.f;
  }
  s_src[n] = ss;
  s_dst[n] = sd;
  denom[n] = 0.f;
}

// ============================================================
// Kernel 3: per-edge e = leaky_relu(s_src[src] + s_dst[dst]); global max
// ============================================================
__global__ __launch_bounds__(256) void gat_edge_e(const int* __restrict__ src,
                                                  const int* __restrict__ dst,
                                                  const float* __restrict__ s_src,
                                                  const float* __restrict__ s_dst,
                                                  float* __restrict__ e_out,
                                                  unsigned* __restrict__ gmax,
                                                  int E) {
  int i = blockIdx.x * blockDim.x + threadIdx.x;
  float e = -INFINITY;
  if (i < E) {
    float v = s_src[src[i]] + s_dst[dst[i]];
    e = (v >= 0.f) ? v : NEG_SLOPE * v;
    e_out[i] = e;
  }
  __shared__ float red[256];
  red[threadIdx.x] = e;
  __syncthreads();
#pragma unroll
  for (int s = 128; s > 0; s >>= 1) {
    if (threadIdx.x < (unsigned)s) red[threadIdx.x] = fmaxf(red[threadIdx.x], red[threadIdx.x + s]);
    __syncthreads();
  }
  if (threadIdx.x == 0) atomicMax(gmax, enc_f32(red[0]));
}

// ============================================================
// Kernel 4: exp_e = exp(e - gmax) (in place); denom[dst] += exp_e
// ============================================================
__global__ __launch_bounds__(256) void gat_edge_exp(const int* __restrict__ dst,
                                                    float* __restrict__ e,      // in: e, out: exp_e
                                                    const unsigned* __restrict__ gmax,
                                                    float* __restrict__ denom,
                                                    int E) {
  int i = blockIdx.x * blockDim.x + threadIdx.x;
  if (i >= E) return;
  float m = dec_f32(*gmax);
  float ex = expf(e[i] - m);
  e[i] = ex;
  atomicAddF32(&denom[dst[i]], ex);
}

// ============================================================
// Kernel 5: out[dst] += alpha * H[src]; 16 threads per edge, float4 each
// ============================================================
__global__ __launch_bounds__(256) void gat_scatter(const int* __restrict__ src,
                                                   const int* __restrict__ dst,
                                                   const float* __restrict__ H,
                                                   const float* __restrict__ expe,
                                                   const float* __restrict__ denom,
                                                   float* __restrict__ out,
                                                   int E) {
  int t = blockIdx.x * blockDim.x + threadIdx.x;
  int i = t >> 4;
  if (i >= E) return;
  int q = (t & 15) << 2;
  int d = dst[i];
  float alpha = expe[i] / (denom[d] + 1e-12f);
  const float4 hv = *(const float4*)(H + (size_t)src[i] * OUTDIM + q);
  float* o = out + (size_t)d * OUTDIM + q;
  atomicAddF32(o + 0, hv.x * alpha);
  atomicAddF32(o + 1, hv.y * alpha);
  atomicAddF32(o + 2, hv.z * alpha);
  atomicAddF32(o + 3, hv.w * alpha);
}

// ============================================================
extern "C" void kernel_launch(void* const* d_in, const int* in_sizes, int n_in,
                              void* d_out, int out_size, void* d_ws, size_t ws_size,
                              hipStream_t stream) {
  const float* X     = (const float*)d_in[0];
  const int*   eidx  = (const int*)d_in[1];
  const float* W     = (const float*)d_in[3];
  const float* a_src = (const float*)d_in[4];
  const float* a_dst = (const float*)d_in[5];
  float* out = (float*)d_out;

  const int N = in_sizes[0] / INDIM;   // 50000
  const int E = in_sizes[1] / 2;       // 800000
  const int* src = eidx;
  const int* dst = eidx + E;

  // workspace carve-out (256B aligned)
  char* ws = (char*)d_ws;
  size_t off = 0;
  auto carve = [&](size_t bytes) -> char* {
    char* p = ws + off;
    off = (off + bytes + 255) & ~(size_t)255;
    return p;
  };
  float*    H     = (float*)carve((size_t)N * OUTDIM * sizeof(float));
  float*    s_src = (float*)carve((size_t)N * sizeof(float));
  float*    s_dst = (float*)carve((size_t)N * sizeof(float));
  float*    denom = (float*)carve((size_t)N * sizeof(float));
  float*    ebuf  = (float*)carve((size_t)E * sizeof(float));
  unsigned* gmax  = (unsigned*)carve(sizeof(unsigned));

  const int nRowTiles = N / 16;                       // 3125
  const int gemmBlocks = (nRowTiles + 7) / 8;         // 8 waves per block
  gat_gemm<<<gemmBlocks, 256, 0, stream>>>(X, W, H, nRowTiles);

  gat_scores<<<(N + 255) / 256, 256, 0, stream>>>(H, a_src, a_dst, s_src, s_dst,
                                                  denom, out, gmax, N);

  gat_edge_e<<<(E + 255) / 256, 256, 0, stream>>>(src, dst, s_src, s_dst, ebuf, gmax, E);

  gat_edge_exp<<<(E + 255) / 256, 256, 0, stream>>>(dst, ebuf, gmax, denom, E);

  const long long scatterThreads = (long long)E * 16;
  gat_scatter<<<(int)((scatterThreads + 255) / 256), 256, 0, stream>>>(src, dst, H, ebuf,
                                                                       denom, out, E);
}